// ES_10917806867103
// MI455X (gfx1250) — compile-verified
//
#include <hip/hip_runtime.h>
#include <hip/hip_bf16.h>
#include <stdint.h>

#define N_SERIES   65536
#define N_TIME     1024
#define SEAS       24
#define LEV_COLS   1024
#define SEA_COLS   1048              // 25 + 1023
#define TT         24                // time tile == ring period
#define NFULL      42                // full tiles: t = 1..1008
#define TAIL       15                // tail tile:  t = 1009..1023
#define YSTRIDE    25                // LDS stride per lane for y tiles (gcd(25,64)=1 -> conflict-free)
#define OSTRIDE    33                // LDS stride for output staging (gcd(33,64)=1)
#define WAVES_PER_BLOCK 4
#define TPB        (WAVES_PER_BLOCK * 32)

__device__ __forceinline__ float fast_rcp(float x) {
#if __has_builtin(__builtin_amdgcn_rcpf)
    return __builtin_amdgcn_rcpf(x);
#else
    return 1.0f / x;
#endif
}

// Issue one time-tile as CNT async global->LDS gathers. LDS layout is [lane][t]
// so the per-step stride (4B) matches the global stride, letting every
// instruction share the same two register operands and differ only in the
// offset immediate (ISA adds INST_OFFSET to BOTH the LDS and global address).
template<int CNT>
__device__ __forceinline__ void issue_tile(uint32_t lds_lane_base, const float* gbase) {
    unsigned long long ga = (unsigned long long)(uintptr_t)gbase;
#pragma unroll
    for (int k = 0; k < CNT; ++k) {
        asm volatile("global_load_async_to_lds_b32 %0, %1, off offset:%c2"
                     :: "v"(lds_lane_base), "v"(ga), "i"(k * 4) : "memory");
    }
}

// Scan KCNT steps of one tile. Ring buffer r[] stays in VGPRs because tiles are
// aligned to the 24-long seasonality period (static ring index == k).
template<int KCNT>
__device__ __forceinline__ void compute_tile(const float* yt_lane,   // &s_y[w][buf][lane*YSTRIDE]
                                             float* lv_lane,         // &s_lv[w][lane]
                                             float* se_lane,         // &s_se[w][lane]
                                             float (&r)[SEAS], float& lev,
                                             float a, float b, float oma, float omb) {
#pragma unroll
    for (int k = 0; k < KCNT; ++k) {
        const float yt = yt_lane[k];
        const float st = r[k];
        const float nl = __builtin_fmaf(a, yt * fast_rcp(st), oma * lev);
        const float ns = __builtin_fmaf(b, yt * fast_rcp(nl), omb * st);
        r[k] = ns;
        lev  = nl;
        lv_lane[k * OSTRIDE] = nl;   // staged [time k][series lane]
        se_lane[k * OSTRIDE] = ns;
    }
}

// Coalesced writeback: lane indexes time (KCNT valid lanes), loop walks the 32
// series rows; nontemporal stores keep the write-once outputs out of L2.
template<int KCNT>
__device__ __forceinline__ void writeback_tile(const float* lv_t, const float* se_t,
                                               float* __restrict__ out_lev,
                                               float* __restrict__ out_sea,
                                               int sbase, int tbase, int lane) {
    if (lane < KCNT) {
        const int tcol = tbase + lane;
        const float* lq = lv_t + lane * OSTRIDE;
        const float* sq = se_t + lane * OSTRIDE;
        float* po = out_lev + (size_t)sbase * LEV_COLS + tcol;
        float* ps = out_sea + (size_t)sbase * SEA_COLS + 24 + tcol;
#pragma unroll 4
        for (int s = 0; s < 32; ++s) {
            __builtin_nontemporal_store(lq[s], po);
            __builtin_nontemporal_store(sq[s], ps);
            po += LEV_COLS;
            ps += SEA_COLS;
        }
    }
}

__global__ __launch_bounds__(TPB) void es_scan_kernel(
    const float* __restrict__ y,
    const int*   __restrict__ idxs,
    const float* __restrict__ lev_sms,
    const float* __restrict__ seas_sms,
    const float* __restrict__ init_seas,
    float* __restrict__ out_lev,
    float* __restrict__ out_sea)
{
    // Per-wave private LDS regions; waves never touch each other's region,
    // so no workgroup barriers anywhere (wave-synchronous + DScnt/ASYNCcnt).
    __shared__ float s_y [WAVES_PER_BLOCK][2][32 * YSTRIDE];  // DMA'd y tiles, [lane][t]
    __shared__ float s_lv[WAVES_PER_BLOCK][TT * OSTRIDE];     // staged levels  [t][lane]
    __shared__ float s_se[WAVES_PER_BLOCK][TT * OSTRIDE];     // staged seasons [t][lane]

    const int w      = threadIdx.x >> 5;
    const int lane   = threadIdx.x & 31;
    const int sbase  = blockIdx.x * TPB + w * 32;
    const int series = sbase + lane;

    // ---------------- init phase ----------------
    const int   idx = idxs[series];
    const float a   = 1.0f / (1.0f + __expf(-lev_sms[idx]));
    const float b   = 1.0f / (1.0f + __expf(-seas_sms[idx]));
    const float oma = 1.0f - a;
    const float omb = 1.0f - b;

    float r[SEAS];
    float s00;
    {
        const float* ip = init_seas + (size_t)idx * SEAS;
        float s0[SEAS];
#pragma unroll
        for (int j = 0; j < SEAS; ++j) s0[j] = __expf(ip[j]);
        s00 = s0[0];
#pragma unroll
        for (int j = 0; j < SEAS - 1; ++j) r[j] = s0[j + 1];   // buf0 = rotate-left(s0)
        r[SEAS - 1] = s0[0];

        float* sp = out_sea + (size_t)series * SEA_COLS;       // season cols 0..24
#pragma unroll
        for (int j = 0; j < SEAS; ++j) __builtin_nontemporal_store(s0[j], sp + j);
        __builtin_nontemporal_store(s0[0], sp + SEAS);
    }
    float lev = y[(size_t)series * N_TIME] * fast_rcp(s00);
    __builtin_nontemporal_store(lev, out_lev + (size_t)series * LEV_COLS);

    // ---------------- DMA double-buffer setup ----------------
    const float* ysrc = y + (size_t)series * N_TIME;
    const uint32_t lb0 = (uint32_t)(uintptr_t)(&s_y[w][0][0]) + (uint32_t)(lane * YSTRIDE * 4);
    const uint32_t lb1 = (uint32_t)(uintptr_t)(&s_y[w][1][0]) + (uint32_t)(lane * YSTRIDE * 4);
    const float* yt0 = &s_y[w][0][lane * YSTRIDE];
    const float* yt1 = &s_y[w][1][lane * YSTRIDE];
    float* lvl = &s_lv[w][lane];
    float* sel = &s_se[w][lane];
    const float* lvt = &s_lv[w][0];
    const float* set = &s_se[w][0];

    issue_tile<TT>(lb0, ysrc + 1);                               // tile 0 -> buf 0

    // ---------------- main scan ----------------
    // Tiles 0..40: issue next full tile, wait current, compute, write back.
    for (int c = 0; c < NFULL - 1; ++c) {
        const int n = c + 1;
        issue_tile<TT>((n & 1) ? lb1 : lb0, ysrc + TT * n + 1);
        asm volatile("s_wait_asynccnt 24" ::: "memory");         // tile c landed (in-order)
        compute_tile<TT>((c & 1) ? yt1 : yt0, lvl, sel, r, lev, a, b, oma, omb);
        writeback_tile<TT>(lvt, set, out_lev, out_sea, sbase, TT * c + 1, lane);
    }
    // Tile 41: prefetch the 15-step tail (tile 42 -> buf 0), then finish 41.
    issue_tile<TAIL>(lb0, ysrc + TT * NFULL + 1);
    asm volatile("s_wait_asynccnt 15" ::: "memory");
    compute_tile<TT>(yt1, lvl, sel, r, lev, a, b, oma, omb);     // tile 41 is in buf 1
    writeback_tile<TT>(lvt, set, out_lev, out_sea, sbase, TT * (NFULL - 1) + 1, lane);
    // Tail tile 42: t = 1009..1023.
    asm volatile("s_wait_asynccnt 0" ::: "memory");
    compute_tile<TAIL>(yt0, lvl, sel, r, lev, a, b, oma, omb);
    writeback_tile<TAIL>(lvt, set, out_lev, out_sea, sbase, TT * NFULL + 1, lane);
}

extern "C" void kernel_launch(void* const* d_in, const int* in_sizes, int n_in,
                              void* d_out, int out_size, void* d_ws, size_t ws_size,
                              hipStream_t stream) {
    const float* y         = (const float*)d_in[0];
    const int*   idxs      = (const int*)  d_in[1];
    const float* lev_sms   = (const float*)d_in[2];
    const float* seas_sms  = (const float*)d_in[3];
    const float* init_seas = (const float*)d_in[4];

    float* out_lev = (float*)d_out;
    float* out_sea = out_lev + (size_t)N_SERIES * LEV_COLS;

    dim3 grid(N_SERIES / TPB), block(TPB);
    hipLaunchKernelGGL(es_scan_kernel, grid, block, 0, stream,
                       y, idxs, lev_sms, seas_sms, init_seas, out_lev, out_sea);
}